// CTCLoss_23502061043921
// MI455X (gfx1250) — compile-verified
//
#include <hip/hip_runtime.h>
#include <hip/hip_bf16.h>

// Problem constants (match reference)
#define BB 32
#define TT 1024
#define VV 1024
#define LL 200
#define SSTATES 401            // 2*L + 1
#define NEGF (-1.0e30f)

// ---------------------------------------------------------------------------
// TDM descriptor helper: async DMA of one contiguous 4KB row (1024 x f32)
// from global memory into LDS. D# layout per CDNA5 ISA §8 (CDNA5 ISA).
// ---------------------------------------------------------------------------
typedef __attribute__((ext_vector_type(4))) unsigned int tdm_v4u;
typedef __attribute__((ext_vector_type(8))) int          tdm_v8i;
typedef __attribute__((ext_vector_type(4))) int          tdm_v4i;

__device__ __forceinline__ void tdm_load_row(const float* gsrc, unsigned lds_byte_addr) {
    unsigned long long ga = (unsigned long long)gsrc;
    tdm_v4u g0;
    g0.x = 1u;                                                   // count=1 (valid user D#)
    g0.y = lds_byte_addr;                                        // lds_addr (bytes)
    g0.z = (unsigned)(ga & 0xffffffffull);                       // global_addr[31:0]
    g0.w = (unsigned)((ga >> 32) & 0x1ffffffull) | (2u << 30);   // global_addr[56:32] | type=2
    tdm_v8i g1;
    g1[0] = (int)(2u << 16);          // workgroup_mask=0 | data_size=2 (4 bytes)
    g1[1] = (int)((unsigned)VV << 16);// atomic_barrier_addr=0 | tensor_dim0[15:0]=1024
    g1[2] = (int)(1u << 16);          // tensor_dim0[31:16]=0 | tensor_dim1[15:0]=1
    g1[3] = (int)((unsigned)VV << 16);// tensor_dim1[31:16]=0 | tile_dim0=1024
    g1[4] = 1;                        // tile_dim1=1 | tile_dim2=0
    g1[5] = VV;                       // tensor_dim0_stride[31:0]=1024
    g1[6] = (int)((unsigned)VV << 16);// stride0[47:32]=0 | tensor_dim1_stride[15:0]
    g1[7] = 0;                        // tensor_dim1_stride[47:16]=0
    tdm_v4i g2 = {0, 0, 0, 0};        // dims 2..4 unused (2-D tensor)
    tdm_v4i g3 = {0, 0, 0, 0};
#if defined(__clang_major__) && (__clang_major__ >= 23)
    tdm_v8i g4 = {0, 0, 0, 0, 0, 0, 0, 0};
    __builtin_amdgcn_tensor_load_to_lds(g0, g1, g2, g3, g4, 0);
#else
    __builtin_amdgcn_tensor_load_to_lds(g0, g1, g2, g3, 0);
#endif
}

__device__ __forceinline__ float logaddexp3(float a, float b, float c) {
    float m = fmaxf(a, fmaxf(b, c));
    return m + __logf(__expf(a - m) + __expf(b - m) + __expf(c - m));
}

// ---------------------------------------------------------------------------
// Kernel 1: Z[b,t] = logsumexp_v logits[b,t,v].  One 256-thread block per row.
// Pure HBM-bandwidth pass: 128 MB read -> ~5.5 us at 23.3 TB/s. Leaves the
// whole logits tensor resident in the 192 MB L2 for the DP kernel.
// ---------------------------------------------------------------------------
__global__ __launch_bounds__(256) void ctc_lse_kernel(const float* __restrict__ logits,
                                                      float* __restrict__ Z) {
    const int row = blockIdx.x;                       // 0 .. B*T-1
    const float4* rp = (const float4*)(logits + (size_t)row * VV);
    float4 v = rp[threadIdx.x];                       // coalesced 16B/lane

    float m = fmaxf(fmaxf(v.x, v.y), fmaxf(v.z, v.w));
    #pragma unroll
    for (int off = 16; off > 0; off >>= 1) m = fmaxf(m, __shfl_xor(m, off, 32));

    __shared__ float sm[8];
    __shared__ float ss[8];
    const int wave = threadIdx.x >> 5;
    const int lane = threadIdx.x & 31;
    if (lane == 0) sm[wave] = m;
    __syncthreads();
    float M = sm[0];
    #pragma unroll
    for (int i = 1; i < 8; ++i) M = fmaxf(M, sm[i]);

    float s = __expf(v.x - M) + __expf(v.y - M) + __expf(v.z - M) + __expf(v.w - M);
    #pragma unroll
    for (int off = 16; off > 0; off >>= 1) s += __shfl_xor(s, off, 32);
    if (lane == 0) ss[wave] = s;
    __syncthreads();
    if (threadIdx.x == 0) {
        float tot = ss[0] + ss[1] + ss[2] + ss[3] + ss[4] + ss[5] + ss[6] + ss[7];
        Z[row] = M + __logf(tot);
    }
}

// ---------------------------------------------------------------------------
// Kernel 2: alpha DP. One 512-thread block (16 waves) per batch element.
// Thread s < 401 owns extended state s; alpha shifts go through LDS.
// Logit row t+1 is TDM-prefetched into the other half of a double buffer
// while step t computes: gathers become ds_load from LDS, global traffic
// (from L2) is fully overlapped with the recurrence.
// ---------------------------------------------------------------------------
__global__ __launch_bounds__(512) void ctc_dp_kernel(const float* __restrict__ logits,
                                                     const int* __restrict__ targets,
                                                     const int* __restrict__ loglen,
                                                     const int* __restrict__ tgtlen,
                                                     const float* __restrict__ Z,
                                                     float* __restrict__ out) {
    __shared__ float rowbuf[2][VV];                   // 8 KB double buffer
    __shared__ float ash[SSTATES + 2];                // alpha with 2-entry NEG pad

    const int b   = blockIdx.x;
    const int tid = threadIdx.x;
    const int s   = tid;
    const bool active = (s < SSTATES);

    // Per-state constants: extended label + skip-transition permission
    int lab = 0;            // BLANK
    bool allowed = false;   // s-2 -> s skip allowed
    if (active && (s & 1)) {
        lab = targets[b * LL + (s >> 1)];
        if (s >= 3) allowed = (lab != targets[b * LL + (s >> 1) - 1]);
    }
    const int Tlen = loglen[b];
    const size_t rowbase = (size_t)b * TT * VV;
    const size_t zbase   = (size_t)b * TT;
    const unsigned lds0 = (unsigned)(unsigned long long)(&rowbuf[0][0]);
    const unsigned lds1 = (unsigned)(unsigned long long)(&rowbuf[1][0]);

    // Prologue: prefetch rows 0 and 1; wait for row 0 (in-order completion).
    if (tid == 0) {
        tdm_load_row(logits + rowbase, lds0);
        tdm_load_row(logits + rowbase + VV, lds1);
        __builtin_amdgcn_s_wait_tensorcnt(1);
    }
    if (tid < 2) ash[tid] = NEGF;                     // NEG pad for s-1 / s-2
    __syncthreads();                                   // row 0 + pads visible

    // t = 0 init: alpha0[0] = lp(blank), alpha0[1] = lp(target0), else NEG
    float alpha = NEGF;
    if (active) {
        if (s < 2) alpha = rowbuf[0][lab] - Z[zbase];
        ash[2 + s] = alpha;
    }
    __syncthreads();                                   // also fences reads of rowbuf[0]

    for (int t = 1; t < TT; ++t) {
        if (tid == 0) {
            if (t + 1 < TT) {
                tdm_load_row(logits + rowbase + (size_t)(t + 1) * VV,
                             ((t + 1) & 1) ? lds1 : lds0);
                __builtin_amdgcn_s_wait_tensorcnt(1);  // row t resident (in-order)
            } else {
                __builtin_amdgcn_s_wait_tensorcnt(0);
            }
        }
        __syncthreads();                               // row t + ash(t-1) visible

        if (active) {
            const float a1 = ash[1 + s];               // alpha[s-1]
            const float a2 = allowed ? ash[s] : NEGF;  // alpha[s-2] if skip allowed
            const float lp = rowbuf[t & 1][lab] - Z[zbase + t];
            const float nv = logaddexp3(alpha, a1, a2) + lp;
            if (t < Tlen) alpha = nv;                  // freeze past input length
        }
        __syncthreads();                               // all ash/rowbuf reads done
        if (active) ash[2 + s] = alpha;
    }
    __syncthreads();

    if (tid == 0) {
        const int tl = tgtlen[b];
        const float aL = ash[2 + 2 * tl - 1];
        const float aB = ash[2 + 2 * tl];
        const float m  = fmaxf(aL, aB);
        out[b] = -(m + __logf(__expf(aL - m) + __expf(aB - m)));
    }
}

// ---------------------------------------------------------------------------
extern "C" void kernel_launch(void* const* d_in, const int* in_sizes, int n_in,
                              void* d_out, int out_size, void* d_ws, size_t ws_size,
                              hipStream_t stream) {
    (void)in_sizes; (void)n_in; (void)out_size; (void)ws_size;
    const float* logits  = (const float*)d_in[0];   // [B,T,V] fp32
    const int*   targets = (const int*)d_in[1];     // [B,L]
    const int*   loglen  = (const int*)d_in[2];     // [B]
    const int*   tgtlen  = (const int*)d_in[3];     // [B]
    float*       out     = (float*)d_out;           // [B]
    float*       Z       = (float*)d_ws;            // [B*T] logsumexp scratch (128 KB)

    ctc_lse_kernel<<<dim3(BB * TT), dim3(256), 0, stream>>>(logits, Z);
    ctc_dp_kernel<<<dim3(BB), dim3(512), 0, stream>>>(logits, targets, loglen, tgtlen, Z, out);
}